// MaskedUpsample_26225070309539
// MI455X (gfx1250) — compile-verified
//
#include <hip/hip_runtime.h>

typedef __attribute__((ext_vector_type(2))) float v2f;
typedef __attribute__((ext_vector_type(8))) float v8f;

#define BIGF 1.0e10f

// ---------------------------------------------------------------------------
// Kernel 1: masked nearest-neighbor index via V_WMMA_F32_16X16X4_F32.
// One wave32 owns a tile of 16 queries and sweeps all N2 supports in 16-row
// tiles. Each WMMA computes score[m][n] = |s_m|^2 - 2 <s_m, q_n> directly:
//   A (16x4, per support row m): ( x_s, y_s, z_s, snorm )   snorm=BIG if masked
//   B (4x16, per query col  n): (-2x_q,-2y_q,-2z_q, 1    )
// The |q|^2 term is constant per query column and drops out of the argmin.
//
// A-matrix f32 16x4 layout: lanes 0-15 hold K=0,1 ; lanes 16-31 hold K=2,3
// (M = lane&15).  B mirrored (N = lane&15).  D 16x16 f32: lane (N=lane&15)
// holds rows M = vgpr + (lane>=16 ? 8 : 0) in its 8 VGPRs.
// ---------------------------------------------------------------------------
__global__ __launch_bounds__(256)
void nn_idx_wmma(const float* __restrict__ up_xyz,        // (B,3,N1)
                 const float* __restrict__ xyz,           // (B,3,N2)
                 const unsigned char* __restrict__ mask,  // (B,N2) bool
                 int* __restrict__ idx_out,               // (B,N1)
                 int N1, int N2)
{
  const int wavesPerBlock = blockDim.x >> 5;
  const int wid  = blockIdx.x * wavesPerBlock + ((int)threadIdx.x >> 5);
  const int lane = (int)threadIdx.x & 31;
  const int tilesPerBatch = N1 >> 4;
  const int b  = wid / tilesPerBatch;
  const int qt = wid - b * tilesPerBatch;

  const int  n  = lane & 15;
  const bool hi = lane >= 16;
  const int  q  = (qt << 4) + n;

  // Build the per-lane B operand (held constant for the whole sweep).
  const float* upb = up_xyz + (size_t)b * 3 * N1;
  const float xq = upb[q];
  const float yq = upb[(size_t)N1 + q];
  const float zq = upb[(size_t)2 * N1 + q];

  v2f Bm;
  Bm.x = hi ? (-2.0f * zq) : (-2.0f * xq);
  Bm.y = hi ?  1.0f        : (-2.0f * yq);

  const float*         sb = xyz  + (size_t)b * 3 * N2;
  const unsigned char* mb = mask + (size_t)b * N2;

  float best    = 3.4e38f;
  int   bestIdx = 0;

  const int ntiles = N2 >> 4;
  for (int t = 0; t < ntiles; ++t) {
    const int m = (t << 4) + n;
    const float xs = sb[m];
    const float ys = sb[(size_t)N2 + m];
    const float zs = sb[(size_t)2 * N2 + m];
    // Masked supports -> huge norm term, guaranteed to lose every argmin.
    const float sn = mb[m] ? fmaf(xs, xs, fmaf(ys, ys, zs * zs)) : BIGF;

    v2f Am;
    Am.x = hi ? zs : xs;
    Am.y = hi ? sn : ys;

    v8f Cz = {};
    // 8 args: (neg_a, A, neg_b, B, c_mod, C, reuse_a, reuse_b)
    v8f D = __builtin_amdgcn_wmma_f32_16x16x4_f32(
        false, Am, false, Bm, (short)0, Cz, false, false);

    const int mbase = (t << 4) + (hi ? 8 : 0);
#pragma unroll
    for (int r = 0; r < 8; ++r) {
      float s = D[r];
      if (s < best) { best = s; bestIdx = mbase + r; }
    }
  }

  // Lane L covers supports {tile*16 + 0..7}, lane L+16 covers {tile*16 + 8..15}
  // for the same query column: merge the two halves (wave32 shuffle).
  float obest = __shfl_xor(best, 16, 32);
  int   oidx  = __shfl_xor(bestIdx, 16, 32);
  if (obest < best || (obest == best && oidx < bestIdx)) {
    best = obest; bestIdx = oidx;
  }
  if (!hi) idx_out[(size_t)b * N1 + q] = bestIdx;
}

// ---------------------------------------------------------------------------
// Kernel 2: feature gather  out[b,c,i] = feat[b,c,idx[b,i]].
// 4 channels per thread (amortizes the idx load, which stays L2-resident);
// stores are fully coalesced over the query dimension i.
// ---------------------------------------------------------------------------
__global__ __launch_bounds__(256)
void gather_feat(const float* __restrict__ feat,  // (B,C,N2)
                 const int* __restrict__ idx,     // (B,N1)
                 float* __restrict__ out,         // (B,C,N1)
                 int C, int N1, int N2, long long total)
{
  const int CPT = 4;
  long long t = (long long)blockIdx.x * blockDim.x + threadIdx.x;
  if (t >= total) return;

  const int cGroups = C / CPT;
  int i   = (int)(t % N1);
  long long bc = t / N1;
  int c4  = (int)(bc % cGroups);
  int b   = (int)(bc / cGroups);

  int j = idx[(size_t)b * N1 + i];
  const float* fb = feat + ((size_t)b * C + (size_t)c4 * CPT) * N2 + j;
  float*       ob = out  + ((size_t)b * C + (size_t)c4 * CPT) * N1 + i;
#pragma unroll
  for (int k = 0; k < CPT; ++k) {
    ob[(size_t)k * N1] = fb[(size_t)k * N2];
  }
}

// ---------------------------------------------------------------------------
extern "C" void kernel_launch(void* const* d_in, const int* in_sizes, int n_in,
                              void* d_out, int out_size, void* d_ws, size_t ws_size,
                              hipStream_t stream) {
  (void)in_sizes; (void)n_in; (void)out_size; (void)ws_size;

  constexpr int B = 4, N1 = 16384, N2 = 4096, C = 256;

  const float*         up_xyz = (const float*)d_in[0];          // (B,3,N1)
  const float*         xyz    = (const float*)d_in[1];          // (B,3,N2)
  // d_in[2] = up_mask: unused by the reference computation.
  const unsigned char* mask   = (const unsigned char*)d_in[3];  // (B,N2)
  const float*         feat   = (const float*)d_in[4];          // (B,C,N2)
  float*               out    = (float*)d_out;                  // (B,C,N1)
  int*                 idx    = (int*)d_ws;                     // (B,N1) ints

  // Phase 1: 1 wave per 16-query tile -> B*N1/16 = 4096 waves exactly.
  // 256 threads = 8 waves/block -> 512 blocks, no partial waves (EXEC all-1s).
  const int waves  = B * (N1 / 16);
  const int blocks = waves / 8;
  nn_idx_wmma<<<blocks, 256, 0, stream>>>(up_xyz, xyz, mask, idx, N1, N2);

  // Phase 2: coalesced gather, 4 channels per thread.
  const long long total = (long long)B * (C / 4) * N1;
  const int gblocks = (int)((total + 255) / 256);
  gather_feat<<<gblocks, 256, 0, stream>>>(feat, idx, out, C, N1, N2, total);
}